// BoxTransformerEncoder_42640435315248
// MI455X (gfx1250) — compile-verified
//
#include <hip/hip_runtime.h>
#include <math.h>

#define B_ 8
#define N_ 20000
#define C_ 256
#define D_ 256
#define NC_ 91
#define Q_ 900
#define LS 260            // LDS row stride (floats): 260 % 64 = 4 -> conflict-free A reads
#define CAP 2048          // top-k candidate/sort capacity
#define TWO_PI 6.283185307179586f
#define LOG1E4_64 0.14391156831212788f  // ln(10000)/64

typedef __attribute__((ext_vector_type(2))) float v2f;
typedef __attribute__((ext_vector_type(8))) float v8f;
typedef __attribute__((ext_vector_type(4))) unsigned int v4u;
typedef __attribute__((ext_vector_type(4))) int v4i;
typedef __attribute__((ext_vector_type(8))) int v8i;

#if defined(__gfx1250__) && __has_builtin(__builtin_amdgcn_tensor_load_to_lds) && \
    __has_builtin(__builtin_amdgcn_s_wait_tensorcnt)
#define USE_TDM 1
#else
#define USE_TDM 0
#endif

__device__ __forceinline__ v8f wmma4(v2f a, v2f b, v8f c) {
  // V_WMMA_F32_16X16X4_F32 : D = A(16x4) * B(4x16) + C(16x16)
  return __builtin_amdgcn_wmma_f32_16x16x4_f32(false, a, false, b, (short)0, c,
                                               false, false);
}

__device__ __forceinline__ float wsum(float v) {
  for (int o = 16; o; o >>= 1) v += __shfl_xor(v, o, 32);
  return v;
}
__device__ __forceinline__ float wmaxr(float v) {
  for (int o = 16; o; o >>= 1) v = fmaxf(v, __shfl_xor(v, o, 32));
  return v;
}
__device__ __forceinline__ float inv_sig(float x) {
  x = fminf(fmaxf(x, 0.f), 1.f);
  return logf(fmaxf(x, 1e-5f) / fmaxf(1.f - x, 1e-5f));
}
__device__ __forceinline__ float sigm(float x) { return 1.f / (1.f + expf(-x)); }

// One wave computes a 16x16 tile of X(16x256 in LDS, stride LS) @ W(256 x ldW).
// Branch-free: all columns valid.
__device__ __forceinline__ v8f gemm_tile_full(const float* __restrict__ Xl,
                                              const float* __restrict__ W,
                                              int ldW, int colBase, int lane) {
  v8f acc = {};
  const int row  = lane & 15;
  const int koff = (lane >> 4) << 1;          // lanes 0-15: K=0,1 ; 16-31: K=2,3
  const int col  = colBase + (lane & 15);
  const float* xr = Xl + row * LS + koff;
  const float* wp = W + koff * ldW + col;
#pragma unroll 16
  for (int kb = 0; kb < C_; kb += 4) {
    v2f a = *(const v2f*)(xr + kb);           // ds_load_b64, conflict-free
    v2f b;
    b.x = wp[kb * ldW];
    b.y = wp[kb * ldW + ldW];
    acc = wmma4(a, b, acc);
  }
  return acc;
}

// Clamped variant (for 91-wide cls GEMM): out-of-range lanes read column
// ncols-1 (in-bounds); their accumulators are dead values, never consumed.
__device__ __forceinline__ v8f gemm_tile_clamp(const float* __restrict__ Xl,
                                               const float* __restrict__ W,
                                               int ldW, int colBase, int lane,
                                               int ncols) {
  v8f acc = {};
  const int row  = lane & 15;
  const int koff = (lane >> 4) << 1;
  const int col  = colBase + (lane & 15);
  const int colc = (col < ncols) ? col : (ncols - 1);
  const float* xr = Xl + row * LS + koff;
  const float* wp = W + koff * ldW + colc;
#pragma unroll 16
  for (int kb = 0; kb < C_; kb += 4) {
    v2f a = *(const v2f*)(xr + kb);
    v2f b;
    b.x = wp[kb * ldW];
    b.y = wp[kb * ldW + ldW];
    acc = wmma4(a, b, acc);
  }
  return acc;
}

#if USE_TDM
// Issue one TDM descriptor: contiguous `nElem` fp32 from `gaddr` into LDS at
// `ldsOff`, inserting 4 pad dwords after every 256 dwords (row stride 260).
__device__ __forceinline__ void tdm_load_padded(unsigned ldsOff,
                                                unsigned long long ga,
                                                int nElem) {
  v4u g0;
  g0.x = 1u;                                   // count=1, no gather
  g0.y = ldsOff;                               // lds_addr (bytes)
  g0.z = (unsigned)(ga & 0xFFFFFFFFull);       // global_addr[31:0]
  g0.w = (unsigned)((ga >> 32) & 0x1FFFFFFull) | 0x80000000u;  // [56:32] | type=2
  v8i g1;
  // data_size=4B (2<<16) | pad_enable (1<<20) | pad_interval=256dw (7<<22)
  // | pad_amount=4dw (3<<25)
  g1.s0 = (int)((2u << 16) | (1u << 20) | (7u << 22) | (3u << 25));
  g1.s1 = 0;                                   // abar addr / tensor_dim0 lo16 (=0)
  g1.s2 = (int)(0x4000u | (1u << 16));         // tensor_dim0=2^30 hi16, tensor_dim1=1
  g1.s3 = (int)((unsigned)nElem << 16);        // tile_dim0 = nElem (1-D tile)
  g1.s4 = 0;                                   // tile_dim1=0, tile_dim2=0
  g1.s5 = nElem;                               // tensor_dim0_stride lo32
  g1.s6 = 0;
  g1.s7 = 0;
  v4i gz = {0, 0, 0, 0};
#if defined(__clang_major__) && __clang_major__ >= 23
  v8i gz8 = {0, 0, 0, 0, 0, 0, 0, 0};
  __builtin_amdgcn_tensor_load_to_lds(g0, g1, gz, gz, gz8, 0);
#else
  __builtin_amdgcn_tensor_load_to_lds(g0, g1, gz, gz, 0);
#endif
}
#endif

// ---------------------------------------------------------------- kernel 1
// Per 16-token tile: mask+writeback of `output`, cls GEMM (+softmax class-0
// scores), and fused 3-layer bbox MLP. Masking fused here (no separate pass).
__global__ __launch_bounds__(256) void k_tokens(
    const float* __restrict__ src, const int* __restrict__ mask,
    const float* __restrict__ refw,
    const float* __restrict__ Wcls, const float* __restrict__ bcls,
    const float* __restrict__ Wb1, const float* __restrict__ bb1,
    const float* __restrict__ Wb2, const float* __restrict__ bb2,
    const float* __restrict__ Wb3, const float* __restrict__ bb3,
    float* __restrict__ outp, float* __restrict__ enc_cls,
    float* __restrict__ enc_box, float* __restrict__ scores) {
  __shared__ float Xs[16 * LS];
  __shared__ float Hs[16 * LS];
  __shared__ float Cl[16 * 100];
  __shared__ int msk[16];
  const int t = threadIdx.x, lane = t & 31, wave = t >> 5;
  const long rowBase = (long)blockIdx.x * 16;

#if USE_TDM
  // Tensor Data Mover: 16x256 fp32 tile of raw src -> LDS, stride 260.
  if (wave == 0) {
    tdm_load_padded((unsigned)(unsigned long long)(const void*)Xs,
                    (unsigned long long)(const void*)(src + rowBase * C_),
                    16 * C_);
  }
  if (t < 16) msk[t] = mask[rowBase + t];
  if (wave == 0) __builtin_amdgcn_s_wait_tensorcnt(0);
  __syncthreads();
  // apply mask in LDS and write `output` tile back to global
  for (int i = t; i < 16 * 64; i += 256) {
    int r = i >> 6, cc = (i & 63) << 2;
    float4 v;
    if (msk[r]) {
      v.x = 0.f; v.y = 0.f; v.z = 0.f; v.w = 0.f;
      *(float4*)&Xs[r * LS + cc] = v;
    } else {
      v = *(const float4*)&Xs[r * LS + cc];
    }
    *(float4*)(outp + (rowBase + r) * C_ + cc) = v;
  }
#else
  if (t < 16) msk[t] = mask[rowBase + t];
  __syncthreads();
  for (int i = t; i < 16 * 64; i += 256) {
    int r = i >> 6, cc = (i & 63) << 2;
    float4 v = *(const float4*)(src + (rowBase + r) * C_ + cc);
    if (msk[r]) { v.x = 0.f; v.y = 0.f; v.z = 0.f; v.w = 0.f; }
    *(float4*)&Xs[r * LS + cc] = v;
    *(float4*)(outp + (rowBase + r) * C_ + cc) = v;
  }
#endif
  __syncthreads();

  // cls logits: 6 tiles (96 cols, 91 valid) on waves 0..5
  if (wave < 6) {
    v8f acc = gemm_tile_clamp(Xs, Wcls, NC_, wave * 16, lane, NC_);
    int col = wave * 16 + (lane & 15), rb = (lane >> 4) * 8;
    float bias = (col < NC_) ? bcls[col] : 0.f;
    for (int r = 0; r < 8; r++) {
      float v = acc[r] + bias;
      Cl[(rb + r) * 100 + col] = v;
      if (col < NC_) enc_cls[(rowBase + rb + r) * NC_ + col] = v;
    }
  }
  __syncthreads();

  // softmax class-0 prob per row, masked -> scores
  for (int rr = wave * 2; rr < wave * 2 + 2; ++rr) {
    float v1 = Cl[rr * 100 + lane];
    float v2 = Cl[rr * 100 + lane + 32];
    float v3 = (lane < 27) ? Cl[rr * 100 + lane + 64] : -3.0e38f;
    float m = wmaxr(fmaxf(fmaxf(v1, v2), v3));
    float e = expf(v1 - m) + expf(v2 - m) + ((lane < 27) ? expf(v3 - m) : 0.f);
    float s = wsum(e);
    if (lane == 0) {
      float p0 = expf(Cl[rr * 100] - m) / s;
      long gr = rowBase + rr;
      scores[gr] = msk[rr] ? -65504.f : p0;
    }
  }

  // bbox layer 1: h1 = relu(X @ Wb1 + bb1) -> Hs
  for (int tt = wave * 2; tt < wave * 2 + 2; ++tt) {
    v8f acc = gemm_tile_full(Xs, Wb1, C_, tt * 16, lane);
    int col = tt * 16 + (lane & 15), rb = (lane >> 4) * 8;
    float bias = bb1[col];
    for (int r = 0; r < 8; r++) Hs[(rb + r) * LS + col] = fmaxf(acc[r] + bias, 0.f);
  }
  __syncthreads();
  // layer 2: h2 = relu(H @ Wb2 + bb2) -> Xs (X no longer needed)
  for (int tt = wave * 2; tt < wave * 2 + 2; ++tt) {
    v8f acc = gemm_tile_full(Hs, Wb2, C_, tt * 16, lane);
    int col = tt * 16 + (lane & 15), rb = (lane >> 4) * 8;
    float bias = bb2[col];
    for (int r = 0; r < 8; r++) Xs[(rb + r) * LS + col] = fmaxf(acc[r] + bias, 0.f);
  }
  __syncthreads();
  // layer 3 (256->4) + enc_box epilogue
  if (t < 64) {
    int r = t >> 2, c = t & 3;
    float s = bb3[c];
    for (int k = 0; k < C_; k++) s += Xs[r * LS + k] * Wb3[k * 4 + c];
    long gr = rowBase + r;
    enc_box[gr * 4 + c] = sigm(s + inv_sig(refw[gr * 4 + c]));
  }
}

// ---------------------------------------------------------------- kernel 2
// Per-batch deterministic top-Q via 4-bit radix select + bitonic sort.
__global__ __launch_bounds__(256) void k_topk(const float* __restrict__ scores,
                                              int* __restrict__ oidx) {
  __shared__ unsigned cnt[16];
  __shared__ unsigned sh_prefix, sh_rem, nc;
  __shared__ unsigned long long cand[CAP];
  const int t = threadIdx.x;
  const float* s = scores + (long)blockIdx.x * N_;
  if (t == 0) { sh_prefix = 0u; sh_rem = Q_; }
  unsigned pmask = 0u;
  for (int shift = 28; shift >= 0; shift -= 4) {
    if (t < 16) cnt[t] = 0u;
    __syncthreads();
    unsigned pre = sh_prefix;
    for (int n = t; n < N_; n += 256) {
      unsigned u = __float_as_uint(s[n]);
      u = (u & 0x80000000u) ? ~u : (u | 0x80000000u);  // order-preserving map
      if ((u & pmask) == pre) atomicAdd(&cnt[(u >> shift) & 15u], 1u);
    }
    __syncthreads();
    if (t == 0) {
      unsigned rem = sh_rem;
      int d;
      for (d = 15; d > 0; --d) {
        if (cnt[d] >= rem) break;
        rem -= cnt[d];
      }
      sh_rem = rem;
      sh_prefix = pre | ((unsigned)d << shift);
    }
    pmask |= 0xFu << shift;
    __syncthreads();
  }
  const unsigned T = sh_prefix;  // Q-th largest key
  if (t == 0) nc = 0u;
  __syncthreads();
  for (int n = t; n < N_; n += 256) {
    unsigned u = __float_as_uint(s[n]);
    u = (u & 0x80000000u) ? ~u : (u | 0x80000000u);
    if (u >= T) {
      unsigned p = atomicAdd(&nc, 1u);
      if (p < CAP) cand[p] = (((unsigned long long)(~u)) << 32) | (unsigned)n;
    }
  }
  __syncthreads();
  unsigned cn = nc < CAP ? nc : CAP;
  for (unsigned i = cn + t; i < CAP; i += 256) cand[i] = ~0ull;
  // bitonic ascending: (~u, idx) -> value desc, index asc (jax top_k order)
  for (unsigned k = 2; k <= CAP; k <<= 1)
    for (unsigned j = k >> 1; j > 0; j >>= 1) {
      __syncthreads();
      for (unsigned i = t; i < CAP; i += 256) {
        unsigned ixj = i ^ j;
        if (ixj > i) {
          unsigned long long a = cand[i], b = cand[ixj];
          bool up = ((i & k) == 0u);
          if ((a > b) == up) { cand[i] = b; cand[ixj] = a; }
        }
      }
    }
  __syncthreads();
  for (int q = t; q < Q_; q += 256)
    oidx[(long)blockIdx.x * Q_ + q] = (int)(cand[q] & 0xFFFFFFFFull);
}

// ---------------------------------------------------------------- kernel 3
// Gather top-Q embeddings; enc GEMM + LN, bbox MLP -> out_ref, sine -> out_pos.
__global__ __launch_bounds__(256) void k_props(
    const float* __restrict__ outp, const float* __restrict__ refw,
    const int* __restrict__ tidx,
    const float* __restrict__ Wb1, const float* __restrict__ bb1,
    const float* __restrict__ Wb2, const float* __restrict__ bb2,
    const float* __restrict__ Wb3, const float* __restrict__ bb3,
    const float* __restrict__ Wenc, const float* __restrict__ benc,
    const float* __restrict__ lng, const float* __restrict__ lnb,
    float* __restrict__ out_embed, float* __restrict__ out_ref,
    float* __restrict__ out_pos) {
  __shared__ float Xs[16 * LS];
  __shared__ float Hs[16 * LS];
  __shared__ float RW[64];
  __shared__ float RF[64];
  __shared__ long rsrc[16];
  const int t = threadIdx.x, lane = t & 31, wave = t >> 5;
  const long qBase = (long)blockIdx.x * 16;
  if (t < 16) {
    long gq = qBase + t;
    long b = gq / Q_;
    long q = gq - b * Q_;
    rsrc[t] = b * N_ + (long)tidx[b * Q_ + q];
  }
  __syncthreads();
  for (int i = t; i < 16 * 64; i += 256) {
    int r = i >> 6, cc = (i & 63) << 2;
    float4 v = *(const float4*)(outp + rsrc[r] * C_ + cc);
    *(float4*)&Xs[r * LS + cc] = v;
  }
  if (t < 64) RW[t] = refw[rsrc[t >> 2] * 4 + (t & 3)];
  __syncthreads();

  // enc_linear GEMM -> Hs (pre-LayerNorm)
  for (int tt = wave * 2; tt < wave * 2 + 2; ++tt) {
    v8f acc = gemm_tile_full(Xs, Wenc, D_, tt * 16, lane);
    int col = tt * 16 + (lane & 15), rb = (lane >> 4) * 8;
    float bias = benc[col];
    for (int r = 0; r < 8; r++) Hs[(rb + r) * LS + col] = acc[r] + bias;
  }
  __syncthreads();
  // LayerNorm rows -> out_embed
  for (int rr = wave * 2; rr < wave * 2 + 2; ++rr) {
    float x[8]; float sum = 0.f;
    for (int i = 0; i < 8; i++) { x[i] = Hs[rr * LS + lane + 32 * i]; sum += x[i]; }
    float mean = wsum(sum) * (1.f / 256.f);
    float vs = 0.f;
    for (int i = 0; i < 8; i++) { float d = x[i] - mean; vs += d * d; }
    float var = wsum(vs) * (1.f / 256.f);
    float rstd = rsqrtf(var + 1e-5f);
    for (int i = 0; i < 8; i++) {
      int col = lane + 32 * i;
      out_embed[(qBase + rr) * D_ + col] = (x[i] - mean) * rstd * lng[col] + lnb[col];
    }
  }
  __syncthreads();
  // bbox L1: Xs -> Hs
  for (int tt = wave * 2; tt < wave * 2 + 2; ++tt) {
    v8f acc = gemm_tile_full(Xs, Wb1, C_, tt * 16, lane);
    int col = tt * 16 + (lane & 15), rb = (lane >> 4) * 8;
    float bias = bb1[col];
    for (int r = 0; r < 8; r++) Hs[(rb + r) * LS + col] = fmaxf(acc[r] + bias, 0.f);
  }
  __syncthreads();
  // bbox L2: Hs -> Xs
  for (int tt = wave * 2; tt < wave * 2 + 2; ++tt) {
    v8f acc = gemm_tile_full(Hs, Wb2, C_, tt * 16, lane);
    int col = tt * 16 + (lane & 15), rb = (lane >> 4) * 8;
    float bias = bb2[col];
    for (int r = 0; r < 8; r++) Xs[(rb + r) * LS + col] = fmaxf(acc[r] + bias, 0.f);
  }
  __syncthreads();
  // bbox L3 + sigmoid(+inv_sigmoid) -> out_ref (keep in LDS for out_pos)
  if (t < 64) {
    int r = t >> 2, c = t & 3;
    float s = bb3[c];
    for (int k = 0; k < C_; k++) s += Xs[r * LS + k] * Wb3[k * 4 + c];
    float val = sigm(s + inv_sig(RW[t]));
    RF[t] = val;
    out_ref[(qBase + r) * 4 + c] = val;
  }
  __syncthreads();
  // DETR sine embedding: out_pos = sine(ref[:2]) + sine(ref[2:])
  {
    int r = t >> 4;
    int dbase = (t & 15) << 4;
    for (int dd = 0; dd < 16; ++dd) {
      int d = dbase + dd;
      int c = d >> 7;
      int j = (d & 127) >> 1;
      float inv = expf(-(float)j * LOG1E4_64);   // 1 / 10000^(j/64)
      float t1 = RF[r * 4 + c] * TWO_PI * inv;
      float t2 = RF[r * 4 + 2 + c] * TWO_PI * inv;
      out_pos[(qBase + r) * D_ + d] =
          (d & 1) ? (cosf(t1) + cosf(t2)) : (sinf(t1) + sinf(t2));
    }
  }
}

// ---------------------------------------------------------------- launcher
extern "C" void kernel_launch(void* const* d_in, const int* in_sizes, int n_in,
                              void* d_out, int out_size, void* d_ws, size_t ws_size,
                              hipStream_t stream) {
  (void)in_sizes; (void)n_in; (void)out_size; (void)ws_size;
  const float* src  = (const float*)d_in[0];
  const int*   mask = (const int*)d_in[1];
  const float* refw = (const float*)d_in[2];
  const float* Wcls = (const float*)d_in[3];
  const float* bcls = (const float*)d_in[4];
  const float* Wb1  = (const float*)d_in[5];
  const float* bb1  = (const float*)d_in[6];
  const float* Wb2  = (const float*)d_in[7];
  const float* bb2  = (const float*)d_in[8];
  const float* Wb3  = (const float*)d_in[9];
  const float* bb3  = (const float*)d_in[10];
  const float* Wenc = (const float*)d_in[11];
  const float* benc = (const float*)d_in[12];
  const float* lng  = (const float*)d_in[13];
  const float* lnb  = (const float*)d_in[14];

  float* out = (float*)d_out;
  float* o_output = out;                  // (B,N,C)   40,960,000
  float* o_embed  = out + 40960000L;      // (B,Q,D)    1,843,200
  float* o_ref    = out + 42803200L;      // (B,Q,4)       28,800
  float* o_pos    = out + 42832000L;      // (B,Q,D)    1,843,200
  float* o_cls    = out + 44675200L;      // (B,N,NC)  14,560,000
  float* o_box    = out + 59235200L;      // (B,N,4)      640,000

  float* scores = (float*)d_ws;                                    // B*N floats
  int*   tidx   = (int*)((char*)d_ws + (size_t)B_ * N_ * sizeof(float)); // B*Q ints

  k_tokens<<<(B_ * N_) / 16, 256, 0, stream>>>(src, mask, refw, Wcls, bcls,
                                               Wb1, bb1, Wb2, bb2, Wb3, bb3,
                                               o_output, o_cls, o_box, scores);
  k_topk<<<B_, 256, 0, stream>>>(scores, tidx);
  k_props<<<(B_ * Q_) / 16, 256, 0, stream>>>(o_output, refw, tidx, Wb1, bb1,
                                              Wb2, bb2, Wb3, bb3, Wenc, benc,
                                              lng, lnb, o_embed, o_ref, o_pos);
}